// RandAlignGCN_90460601188826
// MI455X (gfx1250) — compile-verified
//
#include <hip/hip_runtime.h>

#define HIDDEN 128
#define NCLS   40
#define ALPHA  0.62f
#define SCAN_B 256

typedef __attribute__((ext_vector_type(2))) float v2f;
typedef __attribute__((ext_vector_type(8))) float v8f;

// ---------------------------------------------------------------- utilities
__global__ void fill_zero_i32_kernel(int* __restrict__ p, long long n) {
  long long i = (long long)blockIdx.x * blockDim.x + threadIdx.x;
  if (i < n) p[i] = 0;
}

__global__ void degree_int_kernel(const int* __restrict__ src, const int* __restrict__ dst,
                                  int* __restrict__ degOi, int* __restrict__ degIi, int nEdges) {
  int e = blockIdx.x * blockDim.x + threadIdx.x;
  if (e < nEdges) {
    atomicAdd(&degOi[src[e]], 1);
    atomicAdd(&degIi[dst[e]], 1);
  }
}

// rsqrt(max(deg,1)); padded tail [n, nPad) gets 0 so padded GEMM rows stay finite
__global__ void degrs_kernel(const int* __restrict__ degOi, const int* __restrict__ degIi,
                             float* __restrict__ degOrs, float* __restrict__ degIrs,
                             int n, int nPad) {
  int i = blockIdx.x * blockDim.x + threadIdx.x;
  if (i < nPad) {
    if (i < n) {
      int co = degOi[i]; if (co < 1) co = 1;
      int ci = degIi[i]; if (ci < 1) ci = 1;
      degOrs[i] = rsqrtf((float)co);
      degIrs[i] = rsqrtf((float)ci);
    } else {
      degOrs[i] = 0.0f;
      degIrs[i] = 0.0f;
    }
  }
}

// ---------------------------------------------------- exclusive prefix scan (3 phases)
__global__ void scan_blocksum_kernel(const int* __restrict__ deg, int* __restrict__ bsum, int n) {
  __shared__ int sh[SCAN_B];
  int i = blockIdx.x * SCAN_B + threadIdx.x;
  sh[threadIdx.x] = (i < n) ? deg[i] : 0;
  __syncthreads();
  for (int s = SCAN_B / 2; s > 0; s >>= 1) {
    if (threadIdx.x < s) sh[threadIdx.x] += sh[threadIdx.x + s];
    __syncthreads();
  }
  if (threadIdx.x == 0) bsum[blockIdx.x] = sh[0];
}

__global__ void scan_serial_kernel(int* __restrict__ bsum, int nb,
                                   int* __restrict__ rowptr, int n) {
  if (blockIdx.x == 0 && threadIdx.x == 0) {
    int run = 0;
    for (int b = 0; b < nb; ++b) { int t = bsum[b]; bsum[b] = run; run += t; }
    rowptr[n] = run;
  }
}

__global__ void scan_final_kernel(const int* __restrict__ deg, const int* __restrict__ bsum,
                                  int* __restrict__ rowptr, int n) {
  __shared__ int sh[SCAN_B];
  int i = blockIdx.x * SCAN_B + threadIdx.x;
  int v = (i < n) ? deg[i] : 0;
  sh[threadIdx.x] = v;
  __syncthreads();
  for (int o = 1; o < SCAN_B; o <<= 1) {             // Hillis-Steele inclusive
    int t = (threadIdx.x >= o) ? sh[threadIdx.x - o] : 0;
    __syncthreads();
    sh[threadIdx.x] += t;
    __syncthreads();
  }
  if (i < n) rowptr[i] = bsum[blockIdx.x] + sh[threadIdx.x] - v;  // exclusive
}

// bucket-fill CSR: esrc[rowptr[dst]+k] = src  (atomics only here and in the degree count)
__global__ void fill_csr_kernel(const int* __restrict__ src, const int* __restrict__ dst,
                                const int* __restrict__ rowptr, int* __restrict__ cursor,
                                int* __restrict__ esrc, int nEdges) {
  int e = blockIdx.x * blockDim.x + threadIdx.x;
  if (e < nEdges) {
    int d = dst[e];
    int p = atomicAdd(&cursor[d], 1);
    esrc[rowptr[d] + p] = src[e];
  }
}

// ------------------------------------------------ GEMM: Y = (X @ W) * rowscale
// fp32 WMMA 16x16x4; block = 16-row M tile, 8 waves = 8 N tiles; stores unguarded
// (output rows padded to nPad). Input rows clamped on the LDS stage.
__global__ __launch_bounds__(256) void gemm128_wmma_kernel(
    const float* __restrict__ X, const float* __restrict__ W,
    const float* __restrict__ rowscale, float* __restrict__ Y, int nRows) {
  __shared__ __align__(16) float As[16 * 132];   // padded stride -> conflict-free
  const int mtile   = blockIdx.x;
  const int wave    = threadIdx.x >> 5;
  const int lane    = threadIdx.x & 31;
  const int rowBase = mtile << 4;

  for (int i = threadIdx.x; i < 16 * HIDDEN; i += 256) {
    int r = i >> 7, c = i & 127;
    int row = rowBase + r; if (row >= nRows) row = nRows - 1;
    As[r * 132 + c] = X[(size_t)row * HIDDEN + c];
  }
  __syncthreads();

  const int mr = lane & 15;
  const int kb = (lane >> 4) << 1;     // A frag: VGPR0=K{0|2}, VGPR1=K{1|3}
  const int n  = (wave << 4) + (lane & 15);

  v8f acc = {0.f, 0.f, 0.f, 0.f, 0.f, 0.f, 0.f, 0.f};
  #pragma unroll 4
  for (int k0 = 0; k0 < HIDDEN; k0 += 4) {
    const float2 av = *(const float2*)&As[mr * 132 + k0 + kb];
    v2f a, b;
    a.x = av.x; a.y = av.y;
    b.x = W[(size_t)(k0 + kb)     * HIDDEN + n];
    b.y = W[(size_t)(k0 + kb + 1) * HIDDEN + n];
    acc = __builtin_amdgcn_wmma_f32_16x16x4_f32(false, a, false, b,
                                                (short)0, acc, false, false);
  }

  const int mBase = rowBase + ((lane >> 4) << 3);   // D: VGPR v -> M=v(+8), N=lane%16
  #pragma unroll
  for (int v = 0; v < 8; ++v) {
    int m = mBase + v;
    Y[(size_t)m * HIDDEN + n] = acc[v] * rowscale[m];
  }
}

// classifier GEMM: Y(nPad x 40) = (X @ W3) * rowscale; 3 waves = 48 padded cols
__global__ __launch_bounds__(96) void gemm_out_wmma_kernel(
    const float* __restrict__ X, const float* __restrict__ W,
    const float* __restrict__ rowscale, float* __restrict__ Y, int nRows) {
  __shared__ __align__(16) float As[16 * 132];
  const int mtile   = blockIdx.x;
  const int wave    = threadIdx.x / 32;
  const int lane    = threadIdx.x & 31;
  const int rowBase = mtile << 4;

  for (int i = threadIdx.x; i < 16 * HIDDEN; i += 96) {
    int r = i >> 7, c = i & 127;
    int row = rowBase + r; if (row >= nRows) row = nRows - 1;
    As[r * 132 + c] = X[(size_t)row * HIDDEN + c];
  }
  __syncthreads();

  const int   mr    = lane & 15;
  const int   kb    = (lane >> 4) << 1;
  const int   n     = (wave << 4) + (lane & 15);
  const int   nc    = (n < NCLS) ? n : (NCLS - 1);
  const float nmask = (n < NCLS) ? 1.0f : 0.0f;

  v8f acc = {0.f, 0.f, 0.f, 0.f, 0.f, 0.f, 0.f, 0.f};
  #pragma unroll 4
  for (int k0 = 0; k0 < HIDDEN; k0 += 4) {
    const float2 av = *(const float2*)&As[mr * 132 + k0 + kb];
    v2f a, b;
    a.x = av.x; a.y = av.y;
    b.x = W[(size_t)(k0 + kb)     * NCLS + nc] * nmask;
    b.y = W[(size_t)(k0 + kb + 1) * NCLS + nc] * nmask;
    acc = __builtin_amdgcn_wmma_f32_16x16x4_f32(false, a, false, b,
                                                (short)0, acc, false, false);
  }

  if (n < NCLS) {
    const int mBase = rowBase + ((lane >> 4) << 3);
    #pragma unroll
    for (int v = 0; v < 8; ++v) {
      int m = mBase + v;
      Y[(size_t)m * NCLS + n] = acc[v] * rowscale[m];
    }
  }
}

// ---- CSR edge accumulation, unrolled x4 for memory-level parallelism:
// 4 independent index loads + 4 global_load_b128 in flight per lane.
__device__ __forceinline__ float4 gather_row_sum(const float* __restrict__ H,
                                                 const int* __restrict__ esrc,
                                                 int beg, int end, int lane) {
  float4 a0 = {0.f, 0.f, 0.f, 0.f};
  float4 a1 = {0.f, 0.f, 0.f, 0.f};
  int j = beg;
  for (; j + 4 <= end; j += 4) {
    int s0 = esrc[j + 0], s1 = esrc[j + 1], s2 = esrc[j + 2], s3 = esrc[j + 3];
    const float4 h0 = *(const float4*)(H + (size_t)s0 * HIDDEN + lane * 4);
    const float4 h1 = *(const float4*)(H + (size_t)s1 * HIDDEN + lane * 4);
    const float4 h2 = *(const float4*)(H + (size_t)s2 * HIDDEN + lane * 4);
    const float4 h3 = *(const float4*)(H + (size_t)s3 * HIDDEN + lane * 4);
    a0.x += h0.x + h1.x; a0.y += h0.y + h1.y; a0.z += h0.z + h1.z; a0.w += h0.w + h1.w;
    a1.x += h2.x + h3.x; a1.y += h2.y + h3.y; a1.z += h2.z + h3.z; a1.w += h2.w + h3.w;
  }
  for (; j < end; ++j) {
    int s = esrc[j];
    const float4 hv = *(const float4*)(H + (size_t)s * HIDDEN + lane * 4);
    a0.x += hv.x; a0.y += hv.y; a0.z += hv.z; a0.w += hv.w;
  }
  float4 r;
  r.x = a0.x + a1.x; r.y = a0.y + a1.y; r.z = a0.z + a1.z; r.w = a0.w + a1.w;
  return r;
}

// gather + fused scale/bias/relu; wave per node, lane owns 4 contiguous feats.
__global__ void gather128_relu_kernel(const float* __restrict__ H,
                                      const int* __restrict__ rowptr,
                                      const int* __restrict__ esrc,
                                      const float* __restrict__ degIrs,
                                      const float* __restrict__ bias,
                                      float* __restrict__ out, int n) {
  long long t = (long long)blockIdx.x * blockDim.x + threadIdx.x;
  int v = (int)(t >> 5);
  if (v >= n) return;
  int lane = (int)(t & 31);
  float4 acc = gather_row_sum(H, esrc, rowptr[v], rowptr[v + 1], lane);
  const float  sc = degIrs[v];
  const float4 bv = *(const float4*)(bias + lane * 4);
  float4 r;
  r.x = fmaxf(acc.x * sc + bv.x, 0.f);
  r.y = fmaxf(acc.y * sc + bv.y, 0.f);
  r.z = fmaxf(acc.z * sc + bv.z, 0.f);
  r.w = fmaxf(acc.w * sc + bv.w, 0.f);
  *(float4*)(out + (size_t)v * HIDDEN + lane * 4) = r;
}

// layer-2 gather fused with RandAlign: h2 = relu(agg*degI+b2);
// h = a*h2 + (1-a)*h1*(||h2||/||h1||), written in place over h1.
__global__ void gather128_randalign_kernel(const float* __restrict__ H,
                                           const int* __restrict__ rowptr,
                                           const int* __restrict__ esrc,
                                           const float* __restrict__ degIrs,
                                           const float* __restrict__ bias,
                                           float* __restrict__ h1io, int n) {
  long long t = (long long)blockIdx.x * blockDim.x + threadIdx.x;
  int v = (int)(t >> 5);
  if (v >= n) return;
  int lane = (int)(t & 31);
  float4 acc = gather_row_sum(H, esrc, rowptr[v], rowptr[v + 1], lane);
  const float  sc = degIrs[v];
  const float4 bv = *(const float4*)(bias + lane * 4);
  float4 h2;
  h2.x = fmaxf(acc.x * sc + bv.x, 0.f);
  h2.y = fmaxf(acc.y * sc + bv.y, 0.f);
  h2.z = fmaxf(acc.z * sc + bv.z, 0.f);
  h2.w = fmaxf(acc.w * sc + bv.w, 0.f);
  const float4 h1 = *(const float4*)(h1io + (size_t)v * HIDDEN + lane * 4);
  float s1 = h1.x * h1.x + h1.y * h1.y + h1.z * h1.z + h1.w * h1.w;
  float s2 = h2.x * h2.x + h2.y * h2.y + h2.z * h2.z + h2.w * h2.w;
  #pragma unroll
  for (int o = 16; o > 0; o >>= 1) {    // wave32 full-row reduction
    s1 += __shfl_xor(s1, o, 32);
    s2 += __shfl_xor(s2, o, 32);
  }
  float ratio = sqrtf(s2) / sqrtf(s1);  // matches reference (no epsilon)
  float w1 = (1.0f - ALPHA) * ratio;
  float4 r;
  r.x = ALPHA * h2.x + w1 * h1.x;
  r.y = ALPHA * h2.y + w1 * h1.y;
  r.z = ALPHA * h2.z + w1 * h1.z;
  r.w = ALPHA * h2.w + w1 * h1.w;
  *(float4*)(h1io + (size_t)v * HIDDEN + lane * 4) = r;
}

// final gather (40-wide) fused with scale + bias, straight into d_out.
__global__ void gather40_kernel(const float* __restrict__ H,
                                const int* __restrict__ rowptr,
                                const int* __restrict__ esrc,
                                const float* __restrict__ degIrs,
                                const float* __restrict__ bias,
                                float* __restrict__ out, int n) {
  long long t = (long long)blockIdx.x * blockDim.x + threadIdx.x;
  int v = (int)(t >> 5);
  if (v >= n) return;
  int lane = (int)(t & 31);
  int f2 = 32 + (lane & 7);                 // lanes 0-7 own cols 32..39
  int beg = rowptr[v], end = rowptr[v + 1];
  float a0 = 0.f, a1 = 0.f, c0 = 0.f, c1 = 0.f;
  int j = beg;
  for (; j + 2 <= end; j += 2) {
    int s0 = esrc[j], s1 = esrc[j + 1];
    float p0 = H[(size_t)s0 * NCLS + lane];
    float q0 = H[(size_t)s0 * NCLS + f2];
    float p1 = H[(size_t)s1 * NCLS + lane];
    float q1 = H[(size_t)s1 * NCLS + f2];
    a0 += p0; c0 += q0; a1 += p1; c1 += q1;
  }
  for (; j < end; ++j) {
    int s = esrc[j];
    a0 += H[(size_t)s * NCLS + lane];
    c0 += H[(size_t)s * NCLS + f2];
  }
  a0 += a1; c0 += c1;
  const float sc = degIrs[v];
  out[(size_t)v * NCLS + lane] = a0 * sc + bias[lane];
  if (lane < 8) out[(size_t)v * NCLS + f2] = c0 * sc + bias[f2];
}

// ---------------------------------------------------------------------------
extern "C" void kernel_launch(void* const* d_in, const int* in_sizes, int n_in,
                              void* d_out, int out_size, void* d_ws, size_t ws_size,
                              hipStream_t stream) {
  const float* X  = (const float*)d_in[0];
  const int*   src= (const int*)  d_in[1];
  const int*   dst= (const int*)  d_in[2];
  const float* W1 = (const float*)d_in[3];
  const float* b1 = (const float*)d_in[4];
  const float* W2 = (const float*)d_in[5];
  const float* b2 = (const float*)d_in[6];
  const float* W3 = (const float*)d_in[7];
  const float* b3 = (const float*)d_in[8];
  float* out = (float*)d_out;

  const int nNodes = in_sizes[0] / HIDDEN;
  const int nEdges = in_sizes[1];
  const int mtiles = (nNodes + 15) / 16;
  const int nPad   = mtiles * 16;
  const int nb     = (nNodes + SCAN_B - 1) / SCAN_B;

  // ---- workspace layout: ints first, then 16B-aligned floats
  int* iws    = (int*)d_ws;
  int* degOi  = iws;
  int* degIi  = degOi + nNodes;
  int* cursor = degIi + nNodes;
  int* rowptr = cursor + nNodes;            // nNodes+1
  int* bsum   = rowptr + nNodes + 1;        // nb
  int* esrc   = bsum + nb;                  // nEdges
  size_t intCnt = (size_t)3 * nNodes + (nNodes + 1) + nb + nEdges;
  intCnt = (intCnt + 3) & ~(size_t)3;
  float* fws    = (float*)(iws + intCnt);
  float* degOrs = fws;                      // nPad
  float* degIrs = degOrs + nPad;            // nPad
  float* bufA   = degIrs + nPad;            // nPad*128  (XW products / logits-pre)
  float* bufB   = bufA + (size_t)nPad * HIDDEN;  // nPad*128  (h1 / blended h)

  auto cdiv = [](long long a, long long b) { return (int)((a + b - 1) / b); };
  const int gatherBlocks = cdiv((long long)nNodes * 32, 256);

  // ---- degrees + CSR (built once; the only atomics in the pipeline)
  fill_zero_i32_kernel<<<cdiv(3LL * nNodes, 256), 256, 0, stream>>>(degOi, 3LL * nNodes);
  degree_int_kernel<<<cdiv(nEdges, 256), 256, 0, stream>>>(src, dst, degOi, degIi, nEdges);
  degrs_kernel<<<cdiv(nPad, 256), 256, 0, stream>>>(degOi, degIi, degOrs, degIrs, nNodes, nPad);
  scan_blocksum_kernel<<<nb, SCAN_B, 0, stream>>>(degIi, bsum, nNodes);
  scan_serial_kernel<<<1, 1, 0, stream>>>(bsum, nb, rowptr, nNodes);
  scan_final_kernel<<<nb, SCAN_B, 0, stream>>>(degIi, bsum, rowptr, nNodes);
  fill_csr_kernel<<<cdiv(nEdges, 256), 256, 0, stream>>>(src, dst, rowptr, cursor, esrc, nEdges);

  // ---- layer 1: h1 = relu( gather( (X@W1)*degO ) * degI + b1 )
  gemm128_wmma_kernel<<<mtiles, 256, 0, stream>>>(X, W1, degOrs, bufA, nNodes);
  gather128_relu_kernel<<<gatherBlocks, 256, 0, stream>>>(bufA, rowptr, esrc, degIrs, b1, bufB, nNodes);

  // ---- layer 2 + RandAlign (in place over h1)
  gemm128_wmma_kernel<<<mtiles, 256, 0, stream>>>(bufB, W2, degOrs, bufA, nNodes);
  gather128_randalign_kernel<<<gatherBlocks, 256, 0, stream>>>(bufA, rowptr, esrc, degIrs, b2, bufB, nNodes);

  // ---- layer 3: logits = gather( (h@W3)*degO ) * degI + b3
  gemm_out_wmma_kernel<<<mtiles, 96, 0, stream>>>(bufB, W3, degOrs, bufA, nNodes);
  gather40_kernel<<<gatherBlocks, 256, 0, stream>>>(bufA, rowptr, esrc, degIrs, b3, out, nNodes);
}